// LMOptimizer_43602507989829
// MI455X (gfx1250) — compile-verified
//
#include <hip/hip_runtime.h>
#include <hip/hip_bf16.h>

// LM superquadric fit on gfx1250 (MI455X).
// 1 workgroup per (b,p) problem; 8 wave32 per block; JtJ/Jtr/cost accumulated
// with V_WMMA_F32_16X16X4_F32 (A==B symmetric Gram trick, r folded in as col 12).

#define NPAR    12
#define NPTS    2048
#define BDIM    8
#define PDIM    16
#define NSAMP   16
#define NSTEPS  5
#define THREADS 256
#define NWAVE   (THREADS / 32)
#define LAMB0   0.1f
#define EPS_DIAG 1e-4f

typedef float v2f __attribute__((ext_vector_type(2)));
typedef float v8f __attribute__((ext_vector_type(8)));

#define UNR _Pragma("unroll")

// ---------------- Dual numbers: value + 12 tangents (forward AD) ------------
struct D { float v; float g[NPAR]; };

__device__ inline D dconst(float c) { D r; r.v = c; UNR for (int i = 0; i < NPAR; i++) r.g[i] = 0.f; return r; }
__device__ inline D dadd(const D& a, const D& b) { D r; r.v = a.v + b.v; UNR for (int i = 0; i < NPAR; i++) r.g[i] = a.g[i] + b.g[i]; return r; }
__device__ inline D dsub(const D& a, const D& b) { D r; r.v = a.v - b.v; UNR for (int i = 0; i < NPAR; i++) r.g[i] = a.g[i] - b.g[i]; return r; }
__device__ inline D dmul(const D& a, const D& b) { D r; r.v = a.v * b.v; UNR for (int i = 0; i < NPAR; i++) r.g[i] = fmaf(a.v, b.g[i], b.v * a.g[i]); return r; }
__device__ inline D ddiv(const D& a, const D& b) {
  float inv = 1.f / b.v; D r; r.v = a.v * inv;
  UNR for (int i = 0; i < NPAR; i++) r.g[i] = (a.g[i] - r.v * b.g[i]) * inv; return r;
}
__device__ inline D dmulc(const D& a, float c) { D r; r.v = a.v * c; UNR for (int i = 0; i < NPAR; i++) r.g[i] = a.g[i] * c; return r; }
__device__ inline D daddc(const D& a, float c) { D r; r.v = a.v + c; UNR for (int i = 0; i < NPAR; i++) r.g[i] = a.g[i]; return r; }
__device__ inline D dexp(const D& a) { float e = expf(a.v); D r; r.v = e; UNR for (int i = 0; i < NPAR; i++) r.g[i] = e * a.g[i]; return r; }
__device__ inline D dsqrt(const D& a) {
  float s = sqrtf(fmaxf(a.v, 0.f)); float is = (s > 0.f) ? 0.5f / s : 0.f;
  D r; r.v = s; UNR for (int i = 0; i < NPAR; i++) r.g[i] = is * a.g[i]; return r;
}
__device__ inline D dsigmoid(const D& a) {
  float s = 1.f / (1.f + expf(-a.v)); float dd = s * (1.f - s);
  D r; r.v = s; UNR for (int i = 0; i < NPAR; i++) r.g[i] = dd * a.g[i]; return r;
}
__device__ inline D dabs(const D& a) {
  float s = (a.v < 0.f) ? -1.f : 1.f;
  D r; r.v = fabsf(a.v); UNR for (int i = 0; i < NPAR; i++) r.g[i] = s * a.g[i]; return r;
}
__device__ inline D dclampg(const D& a, float lo, float hi) {   // jnp.clip (grad gated)
  float gate = (a.v >= lo && a.v <= hi) ? 1.f : 0.f;
  D r; r.v = fminf(fmaxf(a.v, lo), hi);
  UNR for (int i = 0; i < NPAR; i++) r.g[i] = gate * a.g[i]; return r;
}
__device__ inline D dmaxc(const D& a, float c) {                // jnp.maximum(a, const)
  float gate = (a.v >= c) ? 1.f : 0.f;
  D r; r.v = gate ? a.v : c;
  UNR for (int i = 0; i < NPAR; i++) r.g[i] = gate * a.g[i]; return r;
}
__device__ inline D drecip(const D& a) {
  float inv = 1.f / a.v; float m = -inv * inv;
  D r; r.v = inv; UNR for (int i = 0; i < NPAR; i++) r.g[i] = m * a.g[i]; return r;
}
// safe_pow(x, p) = pow(clip(x,1e-10,1e10), p), both dual
__device__ inline D dsafe_pow(const D& a, const D& p) {
  float xc = fminf(fmaxf(a.v, 1e-10f), 1e10f);
  float gate = (a.v >= 1e-10f && a.v <= 1e10f) ? 1.f : 0.f;
  float lx = logf(xc);
  float y = powf(xc, p.v);
  float dydx = y * p.v / xc * gate;
  float dydp = y * lx;
  D r; r.v = y;
  UNR for (int i = 0; i < NPAR; i++) r.g[i] = fmaf(dydx, a.g[i], dydp * p.g[i]);
  return r;
}
// sign(c)*|c|^m  (const base, dual exponent)
__device__ inline D dpow_cb(float c, const D& m) {
  float sg = (c < 0.f) ? -1.f : 1.f; float ac = fabsf(c);
  float y = sg * powf(ac, m.v);
  float lac = (ac > 0.f) ? logf(ac) : 0.f;
  D r; r.v = y; UNR for (int i = 0; i < NPAR; i++) r.g[i] = y * lac * m.g[i]; return r;
}
// sign(c)*safe_pow(|c|, m)
__device__ inline D dspow_cb(float c, const D& m) {
  float sg = (c < 0.f) ? -1.f : 1.f;
  float ac = fminf(fmaxf(fabsf(c), 1e-10f), 1e10f);
  float y = sg * powf(ac, m.v);
  float lac = logf(ac);
  D r; r.v = y; UNR for (int i = 0; i < NPAR; i++) r.g[i] = y * lac * m.g[i]; return r;
}

// ---------------- float helpers ------------------------------------------
__device__ inline float spowf(float x, float p) { return powf(fminf(fmaxf(x, 1e-10f), 1e10f), p); }
__device__ inline float pcbf(float c, float m)  { float s = (c < 0.f) ? -1.f : 1.f; return s * powf(fabsf(c), m); }
__device__ inline float spcbf(float c, float m) { float s = (c < 0.f) ? -1.f : 1.f; return s * powf(fminf(fmaxf(fabsf(c), 1e-10f), 1e10f), m); }
__device__ inline float ppf(float x) {           // sign(x)*clip(|x|,1e-4,1e6)
  float s = (x < 0.f) ? -1.f : 1.f;
  return s * fminf(fmaxf(fabsf(x), 1e-4f), 1e6f);
}

// ---------------- shared state per problem -------------------------------
#define IR   0    // 9 rotation entries (row-major R[i][j] at IR+3*i+j)
#define ISC  9    // 3 scales
#define IS0  12
#define IS1  13
#define IE1  14   // 1/s1
#define IE10 15   // s1/s0
#define IEI0 16   // 1/s0
#define IEH  17   // -s0/2
#define NPRE 18

struct SharedState {
  D   pre[NPRE];
  unsigned long long minkey[NSAMP];   // (f32 bits of min d2)<<32 | point index
  float sampF[NSAMP][3];
  float H[16][16];                    // JtJ (12x12), g in col 12, sum r^2 at [12][12]
  float J[NWAVE][32][16];             // per-wave row staging tiles
  float prm[NPAR], cand[NPAR];
  float lamb, cost0, costNew;
};

// precompute param-only duals (thread 0)
__device__ void precompute(const float* prm, D* pre) {
  D P[NPAR];
  UNR for (int i = 0; i < NPAR; i++) {
    P[i].v = prm[i];
    UNR for (int j = 0; j < NPAR; j++) P[i].g[j] = (i == j) ? 1.f : 0.f;
  }
  UNR for (int k = 0; k < 3; k++) pre[ISC + k] = dclampg(dexp(P[k]), 1e-6f, 100.f);
  pre[IS0] = daddc(dmulc(dsigmoid(P[3]), 1.8f), 0.1f);
  pre[IS1] = daddc(dmulc(dsigmoid(P[4]), 1.8f), 0.1f);
  D n2 = dadd(dadd(dmul(P[5], P[5]), dmul(P[6], P[6])), dadd(dmul(P[7], P[7]), dmul(P[8], P[8])));
  D nr = dmaxc(dsqrt(n2), 1e-12f);
  D qw = ddiv(P[5], nr), qx = ddiv(P[6], nr), qy = ddiv(P[7], nr), qz = ddiv(P[8], nr);
  D xx = dmul(qx, qx), yy = dmul(qy, qy), zz = dmul(qz, qz);
  D xy = dmul(qx, qy), xz = dmul(qx, qz), yz = dmul(qy, qz);
  D wx = dmul(qw, qx), wy = dmul(qw, qy), wz = dmul(qw, qz);
  pre[IR + 0] = daddc(dmulc(dadd(yy, zz), -2.f), 1.f);
  pre[IR + 1] = dmulc(dsub(xy, wz), 2.f);
  pre[IR + 2] = dmulc(dadd(xz, wy), 2.f);
  pre[IR + 3] = dmulc(dadd(xy, wz), 2.f);
  pre[IR + 4] = daddc(dmulc(dadd(xx, zz), -2.f), 1.f);
  pre[IR + 5] = dmulc(dsub(yz, wx), 2.f);
  pre[IR + 6] = dmulc(dsub(xz, wy), 2.f);
  pre[IR + 7] = dmulc(dadd(yz, wx), 2.f);
  pre[IR + 8] = daddc(dmulc(dadd(xx, yy), -2.f), 1.f);
  pre[IE1]  = drecip(pre[IS1]);
  pre[IE10] = ddiv(pre[IS1], pre[IS0]);
  pre[IEI0] = drecip(pre[IS0]);
  pre[IEH]  = dmulc(pre[IS0], -0.5f);
}

__device__ void samp_float(const SharedState& sh, int j, float out[3]) {
  const float U4[4] = {-3.14159265358979f, -1.57079632679490f, 0.f, 1.57079632679490f};
  const float V4[4] = {-1.57079632679490f, -0.785398163397448f, 0.f, 0.785398163397448f};
  float eta = U4[j & 3], om = V4[j >> 2];
  float s0 = sh.pre[IS0].v, s1 = sh.pre[IS1].v;
  float ge = pcbf(cosf(eta), s0);
  float go = pcbf(cosf(om),  s1);
  float fe = spcbf(sinf(eta), s0);
  float fo = spcbf(sinf(om),  s1);
  out[0] = sh.pre[ISC + 0].v * ge * go;
  out[1] = sh.pre[ISC + 1].v * ge * fo;
  out[2] = sh.pre[ISC + 2].v * fe;
}

__device__ void samp_dual(const SharedState& sh, int j, D sd[3]) {
  const float U4[4] = {-3.14159265358979f, -1.57079632679490f, 0.f, 1.57079632679490f};
  const float V4[4] = {-1.57079632679490f, -0.785398163397448f, 0.f, 0.785398163397448f};
  float eta = U4[j & 3], om = V4[j >> 2];
  D ge = dpow_cb(cosf(eta), sh.pre[IS0]);
  D go = dpow_cb(cosf(om),  sh.pre[IS1]);
  D fe = dspow_cb(sinf(eta), sh.pre[IS0]);
  D fo = dspow_cb(sinf(om),  sh.pre[IS1]);
  sd[0] = dmul(sh.pre[ISC + 0], dmul(ge, go));
  sd[1] = dmul(sh.pre[ISC + 1], dmul(ge, fo));
  sd[2] = dmul(sh.pre[ISC + 2], fe);
}

__device__ void point_out_dual(const SharedState& sh, const float* prm, const float pr[3], D out[3]) {
  D o[3];
  UNR for (int k = 0; k < 3; k++) {
    float v = pr[k] - prm[9 + k];
    float s = (v < 0.f) ? -1.f : 1.f;
    float av = fabsf(v);
    float gate = (av >= 1e-4f && av <= 1e6f) ? 1.f : 0.f;
    o[k].v = s * fminf(fmaxf(av, 1e-4f), 1e6f);
    UNR for (int j = 0; j < NPAR; j++) o[k].g[j] = 0.f;
    o[k].g[9 + k] = -gate;     // d/dt = -1 (gated by clip)
  }
  UNR for (int j = 0; j < 3; j++)
    out[j] = dadd(dadd(dmul(o[0], sh.pre[IR + j]), dmul(o[1], sh.pre[IR + 3 + j])),
                  dmul(o[2], sh.pre[IR + 6 + j]));
}

__device__ void point_out_f(const SharedState& sh, const float* prm, const float pr[3], float out[3]) {
  float o[3];
  UNR for (int k = 0; k < 3; k++) o[k] = ppf(pr[k] - prm[9 + k]);
  UNR for (int j = 0; j < 3; j++)
    out[j] = o[0] * sh.pre[IR + j].v + o[1] * sh.pre[IR + 3 + j].v + o[2] * sh.pre[IR + 6 + j].v;
}

__device__ D radial_dual(const SharedState& sh, const D out[3], float wn) {
  D r2 = dadd(dadd(dmul(out[0], out[0]), dmul(out[1], out[1])), dmul(out[2], out[2]));
  D rn = dclampg(dsqrt(r2), 1e-4f, 1e6f);
  D a0 = ddiv(out[0], sh.pre[ISC + 0]); a0 = dmul(a0, a0);
  D a1 = ddiv(out[1], sh.pre[ISC + 1]); a1 = dmul(a1, a1);
  D a2 = ddiv(out[2], sh.pre[ISC + 2]); a2 = dmul(a2, a2);
  D A  = dadd(dsafe_pow(a0, sh.pre[IE1]), dsafe_pow(a1, sh.pre[IE1]));
  D Bv = dsafe_pow(A, sh.pre[IE10]);
  D Cv = dadd(Bv, dsafe_pow(a2, sh.pre[IEI0]));
  D e  = daddc(dsafe_pow(Cv, sh.pre[IEH]), -1.f);
  return dmulc(dmul(rn, dabs(e)), wn * (1.f / (float)NPTS));
}

__device__ float radial_f(const SharedState& sh, const float o[3], float wn) {
  float r2 = o[0] * o[0] + o[1] * o[1] + o[2] * o[2];
  float rn = fminf(fmaxf(sqrtf(r2), 1e-4f), 1e6f);
  float a0 = o[0] / sh.pre[ISC + 0].v; a0 *= a0;
  float a1 = o[1] / sh.pre[ISC + 1].v; a1 *= a1;
  float a2 = o[2] / sh.pre[ISC + 2].v; a2 *= a2;
  float A  = spowf(a0, sh.pre[IE1].v) + spowf(a1, sh.pre[IE1].v);
  float Bv = spowf(A, sh.pre[IE10].v);
  float Cv = Bv + spowf(a2, sh.pre[IEI0].v);
  float e  = spowf(Cv, sh.pre[IEH].v) - 1.f;
  return rn * fabsf(e) * wn * (1.f / (float)NPTS);
}

__device__ void accum_minkeys(SharedState& sh, const float ov[3], float w, int n) {
  float mask = (w < 0.5f) ? 1.f : 0.f;
  UNR for (int j = 0; j < NSAMP; j++) {
    float dx = fmaf(mask, sh.sampF[j][0] - ov[0], 1e-4f);
    float dy = fmaf(mask, sh.sampF[j][1] - ov[1], 1e-4f);
    float dz = fmaf(mask, sh.sampF[j][2] - ov[2], 1e-4f);
    float d2 = dx * dx + dy * dy + dz * dz;
    unsigned long long key =
        (((unsigned long long)__float_as_uint(d2)) << 32) | (unsigned long long)(unsigned)n;
    atomicMin(&sh.minkey[j], key);
  }
}

// 8 x V_WMMA_F32_16X16X4_F32 over one wave's 32 staged rows (A == B Gram trick).
// A layout (16x4 f32): lanes 0-15 -> M=lane, K=0/1 in v0/v1; lanes 16-31 -> K=2/3.
__device__ inline v8f wmma_tile(const SharedState& sh, int wv, int lane, v8f acc) {
  int m  = lane & 15;
  int hk = (lane >> 4) << 1;       // 0 for low half-wave, 2 for high half-wave
  UNR for (int gi = 0; gi < 8; ++gi) {
    v2f a;
    a.x = sh.J[wv][gi * 4 + hk + 0][m];
    a.y = sh.J[wv][gi * 4 + hk + 1][m];
    acc = __builtin_amdgcn_wmma_f32_16x16x4_f32(false, a, false, a, (short)0, acc, false, false);
  }
  return acc;
}

// ---------------- Jacobian pass: fills sh.H = [JtJ | g ; gT | sum r^2] -----
__device__ void jac_pass(SharedState& sh, const float* ptsB, const float* asgB, const float* prm) {
  int tid = threadIdx.x, wave = tid >> 5, lane = tid & 31;
  if (tid == 0) precompute(prm, sh.pre);
  if (tid < NSAMP) sh.minkey[tid] = ~0ull;
  sh.H[tid >> 4][tid & 15] = 0.f;
  __syncthreads();
  if (tid < NSAMP) samp_float(sh, tid, sh.sampF[tid]);
  __syncthreads();

  v8f acc = {0.f, 0.f, 0.f, 0.f, 0.f, 0.f, 0.f, 0.f};
  for (int it = 0; it < NPTS / THREADS; ++it) {
    int n = it * THREADS + tid;
    float pr[3];
    UNR for (int k = 0; k < 3; k++) pr[k] = ppf(ptsB[n * 3 + k]);
    float w = asgB[n * PDIM];
    D outd[3]; point_out_dual(sh, prm, pr, outd);
    D res = radial_dual(sh, outd, w);
    UNR for (int j = 0; j < NPAR; j++) sh.J[wave][lane][j] = res.g[j];
    sh.J[wave][lane][12] = res.v;
    sh.J[wave][lane][13] = 0.f; sh.J[wave][lane][14] = 0.f; sh.J[wave][lane][15] = 0.f;
    float ov[3] = {outd[0].v, outd[1].v, outd[2].v};
    accum_minkeys(sh, ov, w, n);
    acc = wmma_tile(sh, wave, lane, acc);   // same-wave LDS ordering keeps this safe
  }
  __syncthreads();   // min keys final

  if (wave == 0) {   // 16 distance residual rows (argmin branch derivative)
    float row[16];
    UNR for (int c = 0; c < 16; c++) row[c] = 0.f;
    if (lane < NSAMP) {
      unsigned long long key = sh.minkey[lane];
      int idx = (int)(unsigned)(key & 0xffffffffull);
      float pr[3];
      UNR for (int k = 0; k < 3; k++) pr[k] = ppf(ptsB[idx * 3 + k]);
      float w = asgB[idx * PDIM];
      D outd[3]; point_out_dual(sh, prm, pr, outd);
      D sd[3];  samp_dual(sh, lane, sd);
      float mask = (w < 0.5f) ? 1.f : 0.f;
      D d2 = dconst(0.f);
      UNR for (int k = 0; k < 3; k++) {
        D df = daddc(dmulc(dsub(sd[k], outd[k]), mask), 1e-4f);
        d2 = dadd(d2, dmul(df, df));
      }
      D dist = dmulc(dsqrt(d2), 1.f / (float)NSAMP);
      UNR for (int j = 0; j < NPAR; j++) row[j] = dist.g[j];
      row[12] = dist.v;
    }
    UNR for (int c = 0; c < 16; c++) sh.J[0][lane][c] = row[c];
    acc = wmma_tile(sh, 0, lane, acc);
  }

  { // merge wave accumulators: C/D layout lanes 0-15 -> M=v, lanes 16-31 -> M=v+8
    int nn = lane & 15;
    int mb = (lane >> 4) * 8;
    UNR for (int v = 0; v < 8; v++) atomicAdd(&sh.H[mb + v][nn], acc[v]);
  }
}

// ---------------- value-only pass: candidate cost --------------------------
__device__ void val_pass(SharedState& sh, const float* ptsB, const float* asgB, const float* prm) {
  int tid = threadIdx.x;
  if (tid == 0) { precompute(prm, sh.pre); sh.costNew = 0.f; }
  if (tid < NSAMP) sh.minkey[tid] = ~0ull;
  __syncthreads();
  if (tid < NSAMP) samp_float(sh, tid, sh.sampF[tid]);
  __syncthreads();

  float local = 0.f;
  for (int it = 0; it < NPTS / THREADS; ++it) {
    int n = it * THREADS + tid;
    float pr[3];
    UNR for (int k = 0; k < 3; k++) pr[k] = ppf(ptsB[n * 3 + k]);
    float w = asgB[n * PDIM];
    float ov[3]; point_out_f(sh, prm, pr, ov);
    float res = radial_f(sh, ov, w);
    local = fmaf(res, res, local);
    accum_minkeys(sh, ov, w, n);
  }
  atomicAdd(&sh.costNew, local);
  __syncthreads();
  if (tid < NSAMP) {
    float d2 = __uint_as_float((unsigned)(sh.minkey[tid] >> 32));
    float dist = sqrtf(d2) * (1.f / (float)NSAMP);
    atomicAdd(&sh.costNew, dist * dist);
  }
}

// ---------------- main kernel ----------------------------------------------
__global__ __launch_bounds__(THREADS) void lm_superquadric_kernel(
    const float* __restrict__ scale, const float* __restrict__ shape,
    const float* __restrict__ q, const float* __restrict__ trans,
    const float* __restrict__ assign, const float* __restrict__ points,
    float* __restrict__ out) {
  __shared__ SharedState sh;
  int bp = blockIdx.x;
  int b = bp / PDIM, p = bp % PDIM;
  int tid = threadIdx.x;
  const float* ptsB = points + (size_t)b * NPTS * 3;
  const float* asgB = assign + (size_t)b * NPTS * PDIM + p;

  if (tid == 0) {
    UNR for (int k = 0; k < 3; k++) sh.prm[k] = logf(scale[bp * 3 + k]);
    UNR for (int k = 0; k < 2; k++) {
      float sp = (shape[bp * 2 + k] - 0.1f) / 1.8f;
      sh.prm[3 + k] = logf(sp) - log1pf(-sp);
    }
    UNR for (int k = 0; k < 4; k++) sh.prm[5 + k] = q[bp * 4 + k];
    UNR for (int k = 0; k < 3; k++) sh.prm[9 + k] = trans[bp * 3 + k];
    sh.lamb = LAMB0;
  }
  __syncthreads();

  for (int step = 0; step < NSTEPS; ++step) {
    jac_pass(sh, ptsB, asgB, sh.prm);
    __syncthreads();
    if (tid == 0) {
      sh.cost0 = sh.H[12][12];                 // sum r^2 from WMMA col-12 trick
      float M[12][13];
      for (int i = 0; i < 12; i++) {
        float dg = fmaxf(sh.lamb * sh.H[i][i], EPS_DIAG);
        for (int j = 0; j < 12; j++) M[i][j] = sh.H[i][j];
        M[i][i] += dg;
        M[i][12] = sh.H[i][12];                // g from col 12
      }
      for (int c = 0; c < 12; c++) {           // partial-pivot Gaussian elim
        int piv = c; float mx = fabsf(M[c][c]);
        for (int r = c + 1; r < 12; r++) { float a = fabsf(M[r][c]); if (a > mx) { mx = a; piv = r; } }
        if (piv != c) for (int j = c; j < 13; j++) { float t = M[c][j]; M[c][j] = M[piv][j]; M[piv][j] = t; }
        float inv = 1.f / M[c][c];
        for (int r = c + 1; r < 12; r++) {
          float f = M[r][c] * inv;
          for (int j = c; j < 13; j++) M[r][j] -= f * M[c][j];
        }
      }
      float del[12];
      for (int c = 11; c >= 0; c--) {
        float x = M[c][12];
        for (int j = c + 1; j < 12; j++) x -= M[c][j] * del[j];
        del[c] = x / M[c][c];
      }
      for (int i = 0; i < 12; i++) sh.cand[i] = sh.prm[i] - del[i];
    }
    __syncthreads();
    val_pass(sh, ptsB, asgB, sh.cand);
    __syncthreads();
    if (tid == 0) {
      if (sh.costNew < sh.cost0) {
        for (int i = 0; i < 12; i++) sh.prm[i] = sh.cand[i];
        sh.lamb *= 0.1f;
      } else {
        sh.lamb *= 10.f;
      }
    }
    __syncthreads();
  }
  if (tid < NPAR) out[bp * NPAR + tid] = sh.prm[tid];
}

extern "C" void kernel_launch(void* const* d_in, const int* in_sizes, int n_in,
                              void* d_out, int out_size, void* d_ws, size_t ws_size,
                              hipStream_t stream) {
  (void)in_sizes; (void)n_in; (void)out_size; (void)d_ws; (void)ws_size;
  const float* scale  = (const float*)d_in[0];
  const float* shape  = (const float*)d_in[1];
  const float* q      = (const float*)d_in[2];
  const float* trans  = (const float*)d_in[3];
  const float* assign = (const float*)d_in[4];
  const float* points = (const float*)d_in[5];
  float* out = (float*)d_out;
  lm_superquadric_kernel<<<dim3(BDIM * PDIM), dim3(THREADS), 0, stream>>>(
      scale, shape, q, trans, assign, points, out);
}